// CoAttention_75986561401216
// MI455X (gfx1250) — compile-verified
//
#include <hip/hip_runtime.h>

typedef __attribute__((ext_vector_type(2))) float v2f;
typedef __attribute__((ext_vector_type(4))) float v4f;
typedef __attribute__((ext_vector_type(8))) float v8f;

#define B_  16
#define S_  2048
#define D_  256
#define BM  128
#define BN  128
#define BK  32
#define LDK 36   // BK + 4 dword pad: 144B row stride -> 16B aligned, conflict-free b64 frag reads

// ---------------------------------------------------------------------------
// Pass 1: ru[b,i] = u[b,i,:]·w1 ; cv[b,j] = v[b,j,:]·w2   (one wave per row)
// ---------------------------------------------------------------------------
__global__ __launch_bounds__(256) void coattn_bias_kernel(
    const float* __restrict__ u, const float* __restrict__ v,
    const float* __restrict__ w1, const float* __restrict__ w2,
    float* __restrict__ ru, float* __restrict__ cv)
{
    const int gwave = (int)((blockIdx.x * blockDim.x + threadIdx.x) >> 5);
    const int lane  = threadIdx.x & 31;
    const int nRows = B_ * S_;

    const float* src; const float* wv; float* dst; int row;
    if (gwave < nRows) { src = u; wv = w1; dst = ru; row = gwave; }
    else               { src = v; wv = w2; dst = cv; row = gwave - nRows; }

    const v4f* p = (const v4f*)(src + (size_t)row * D_ + lane * 8);
    const v4f* q = (const v4f*)(wv + lane * 8);
    v4f a0 = p[0], a1 = p[1];
    v4f b0 = q[0], b1 = q[1];
    float acc = a0.x*b0.x + a0.y*b0.y + a0.z*b0.z + a0.w*b0.w
              + a1.x*b1.x + a1.y*b1.y + a1.z*b1.z + a1.w*b1.w;
    #pragma unroll
    for (int off = 16; off > 0; off >>= 1)
        acc += __shfl_xor(acc, off, 32);
    if (lane == 0) dst[row] = acc;
}

// ---------------------------------------------------------------------------
// Pass 2: out[b,i,j] = (u[b,i]*w3)·v[b,j] + ru[b,i] + cv[b,j]
// 128x128 block tile, 8 waves (2x4), each wave 4x2 tiles of 16x16, K chunks of 32.
// ---------------------------------------------------------------------------
__global__ __launch_bounds__(256) void coattn_gemm_kernel(
    const float* __restrict__ u, const float* __restrict__ v,
    const float* __restrict__ w3,
    const float* __restrict__ ru, const float* __restrict__ cv,
    float* __restrict__ out)
{
    __shared__ __attribute__((aligned(16))) float uS[BM * LDK];
    __shared__ __attribute__((aligned(16))) float vS[BN * LDK];

    const int b  = blockIdx.z;
    const int iB = blockIdx.y * BM;
    const int jB = blockIdx.x * BN;

    const int tid   = threadIdx.x;
    const int lane  = tid & 31;
    const int wid   = tid >> 5;
    const int waveM = wid & 1;     // 0..1  -> 64-row stripe
    const int waveN = wid >> 1;    // 0..3  -> 32-col stripe
    const int half  = lane >> 4;   // 0/1 selects K pair (A/B) and +8 rows (C/D)
    const int l16   = lane & 15;

    const size_t ubase = ((size_t)b * S_ + iB) * D_;
    const size_t vbase = ((size_t)b * S_ + jB) * D_;

    // staging map: 256 threads -> 32 rows x 8 v4f columns per pass, 4 passes
    const int sRow = tid >> 3;     // 0..31
    const int sVec = tid & 7;      // 0..7

    v8f c[4][2];
    #pragma unroll
    for (int mt = 0; mt < 4; ++mt)
        #pragma unroll
        for (int nt = 0; nt < 2; ++nt)
            c[mt][nt] = v8f{};

    for (int kc = 0; kc < D_; kc += BK) {
        // global -> LDS stage; fold w3 into the u tile
        v4f w3v = *(const v4f*)(w3 + kc + sVec * 4);
        #pragma unroll
        for (int p = 0; p < 4; ++p) {
            const int r = p * 32 + sRow;
            v4f uu = *(const v4f*)(u + ubase + (size_t)r * D_ + kc + sVec * 4);
            v4f vv = *(const v4f*)(v + vbase + (size_t)r * D_ + kc + sVec * 4);
            uu *= w3v;
            *(v4f*)(uS + r * LDK + sVec * 4) = uu;
            *(v4f*)(vS + r * LDK + sVec * 4) = vv;
        }
        __syncthreads();

        #pragma unroll
        for (int k = 0; k < BK; k += 4) {
            const int kk = k + half * 2;
            v2f aF[4], bF[2];
            #pragma unroll
            for (int mt = 0; mt < 4; ++mt)
                aF[mt] = *(const v2f*)(uS + (waveM * 64 + mt * 16 + l16) * LDK + kk);
            #pragma unroll
            for (int nt = 0; nt < 2; ++nt)
                bF[nt] = *(const v2f*)(vS + (waveN * 32 + nt * 16 + l16) * LDK + kk);
            #pragma unroll
            for (int mt = 0; mt < 4; ++mt)
                #pragma unroll
                for (int nt = 0; nt < 2; ++nt)
                    c[mt][nt] = __builtin_amdgcn_wmma_f32_16x16x4_f32(
                        false, aF[mt], false, bF[nt],
                        (short)0, c[mt][nt], false, false);
        }
        __syncthreads();
    }

    // epilogue: add rank-1 biases, coalesced b32 stores along j
    const float* ruB = ru + (size_t)b * S_ + iB;
    const float* cvB = cv + (size_t)b * S_ + jB;
    #pragma unroll
    for (int nt = 0; nt < 2; ++nt) {
        const int colL = waveN * 32 + nt * 16 + l16;
        const float cvv = cvB[colL];
        #pragma unroll
        for (int mt = 0; mt < 4; ++mt) {
            const int rowBase = waveM * 64 + mt * 16 + half * 8;
            #pragma unroll
            for (int r = 0; r < 8; ++r) {
                const int rowL = rowBase + r;
                const float val = c[mt][nt][r] + ruB[rowL] + cvv;
                out[((size_t)b * S_ + (size_t)(iB + rowL)) * S_ + (jB + colL)] = val;
            }
        }
    }
}

// ---------------------------------------------------------------------------
extern "C" void kernel_launch(void* const* d_in, const int* in_sizes, int n_in,
                              void* d_out, int out_size, void* d_ws, size_t ws_size,
                              hipStream_t stream) {
    (void)in_sizes; (void)n_in; (void)out_size; (void)ws_size;
    const float* u  = (const float*)d_in[0];
    const float* v  = (const float*)d_in[1];
    const float* w1 = (const float*)d_in[2];
    const float* w2 = (const float*)d_in[3];
    const float* w3 = (const float*)d_in[4];
    float* out = (float*)d_out;

    float* ru = (float*)d_ws;            // [B_*S_]
    float* cv = ru + (size_t)B_ * S_;    // [B_*S_]

    // Pass 1: 65536 waves total (32768 for ru + 32768 for cv), 8 waves/block
    const int nWaves  = 2 * B_ * S_;
    const int nBlocks = nWaves / 8;
    coattn_bias_kernel<<<nBlocks, 256, 0, stream>>>(u, v, w1, w2, ru, cv);

    // Pass 2: 16x16 tiles of 128x128 per batch
    dim3 grid(S_ / BN, S_ / BM, B_);
    coattn_gemm_kernel<<<grid, 256, 0, stream>>>(u, v, w3, ru, cv, out);
}